// KA_attention_crossinf_scaling_16655883173899
// MI455X (gfx1250) — compile-verified
//
#include <hip/hip_runtime.h>
#include <hip/hip_bf16.h>

// ---------------------------------------------------------------------------
// Reference: B=16,H=12,P=256,D=64,NF=8.  Output = (q, E@k), E=expm(f_out).
// f_out has rank-1 + column structure => cheap build; expm via scaling &
// squaring (s=10, Taylor-Horner m=8); all matmuls on V_WMMA_F32_16X16X4_F32
// (fp32 required: 10 squarings amplify low-precision input error).
// GEMM: 128x64 C-tile/block, 16x64 strip/wave (4 accumulators), LDS double-
// buffered with GLOBAL_LOAD_ASYNC_TO_LDS_B128 prefetch of the A panel and a
// transposed B panel so every fragment is a single aligned ds_load_b64.
// ---------------------------------------------------------------------------

#define BH     192
#define PP     256
#define DD     64
#define BHP    49152           // B*H*P
#define HPD    3072            // H*P  (scale_base/scale_sp are [1,H,P,D])
#define MATEL  65536           // 256*256
#define TOTEL  12582912        // 192*65536
#define QEL    3145728         // 16*12*256*64

typedef float v2f __attribute__((ext_vector_type(2)));
typedef float v8f __attribute__((ext_vector_type(8)));

// gfx1250 async global->LDS copy (16B per lane, tracked by ASYNCcnt).
// LDS operand: low 32 bits of a generic shared-space pointer == LDS byte offset.
#define ASYNC_G2L_B128(ldsptr, gptr)                                           \
    asm volatile("global_load_async_to_lds_b128 %0, %1, off"                   \
                 :: "v"((unsigned)(unsigned long long)(ldsptr)),               \
                    "v"((unsigned long long)(gptr)) : "memory")
#define WAIT_ASYNC0() asm volatile("s_wait_asynccnt 0x0" ::: "memory")

// ---------------------------------------------------------------------------
// Kernel 1: per-row feature map -> sigmoid row-sum (one 64-thread block/row).
// ---------------------------------------------------------------------------
__global__ __launch_bounds__(64)
void sigrow_kernel(const float* __restrict__ q, const float* __restrict__ k,
                   const float* __restrict__ gridf, const float* __restrict__ W,
                   const float* __restrict__ coef,
                   const float* __restrict__ sbase, const float* __restrict__ ssp,
                   float* __restrict__ sq, float* __restrict__ sk) {
    __shared__ float sx[64];
    __shared__ float red[64];
    int idx  = blockIdx.x;                  // 0 .. 2*BHP-1
    int isK  = (idx >= BHP) ? 1 : 0;
    int row  = isK ? (idx - BHP) : idx;
    const float* src = isK ? k : q;
    int d    = threadIdx.x;

    float x   = src[row * DD + d];
    float sg  = 1.0f / (1.0f + __expf(-x));
    sx[d]     = x * sg;                     // silu(x)
    __syncthreads();

    float base = 0.0f;
#pragma unroll 8
    for (int j = 0; j < DD; ++j) base += sx[j] * W[d * DD + j];   // silu(x) @ W^T

    float fs = 0.0f;
#pragma unroll
    for (int f = 0; f < 8; ++f) fs += coef[d * 8 + f] * __sinf(gridf[f] * x);

    int hp   = row % HPD;
    float v  = fs * ssp[hp * DD + d] + base * sbase[hp * DD + d];
    red[d]   = 1.0f / (1.0f + __expf(-v));  // sigmoid
    __syncthreads();

    if (d == 0) {
        float s = 0.0f;
        for (int j = 0; j < DD; ++j) s += red[j];
        (isK ? sk : sq)[row] = s;
    }
}

// ---------------------------------------------------------------------------
// Kernel 2: skw[bh][j] = sum_j' sk[bh][j'] * lin_w[j][j']   (blocks 0..191)
//           rlw[j]     = sum_j' lin_w[j][j']                (block 192)
// ---------------------------------------------------------------------------
__global__ __launch_bounds__(256)
void skw_kernel(const float* __restrict__ sk, const float* __restrict__ lw,
                float* __restrict__ skw, float* __restrict__ rlw) {
    __shared__ float sks[PP];
    int bh = blockIdx.x;
    int j  = threadIdx.x;
    if (bh < BH) {
        sks[j] = sk[bh * PP + j];
        __syncthreads();
        float acc = 0.0f;
        for (int jp = 0; jp < PP; ++jp) acc += sks[jp] * lw[j * PP + jp];
        skw[bh * PP + j] = acc;
    } else {
        float acc = 0.0f;
        for (int jp = 0; jp < PP; ++jp) acc += lw[j * PP + jp];
        rlw[j] = acc;
    }
}

// ---------------------------------------------------------------------------
// Kernel 3: A[bh][i][j] = 2^-s * ( sq[bh][i]*rlw[j] + skw[bh][j] + lin_b[j] )
// ---------------------------------------------------------------------------
__global__ __launch_bounds__(256)
void buildA_kernel(const float* __restrict__ sq, const float* __restrict__ rlw,
                   const float* __restrict__ skw, const float* __restrict__ lb,
                   float* __restrict__ A, float scaleFac) {
    int idx = blockIdx.x * 256 + threadIdx.x;
    int bh  = idx >> 16;
    int rem = idx & 65535;
    int i   = rem >> 8;
    int j   = rem & 255;
    A[idx]  = scaleFac * (sq[bh * PP + i] * rlw[j] + skw[bh * PP + j] + lb[j]);
}

// Kernel 4: R = I + A/m  (start of Horner)
__global__ __launch_bounds__(256)
void initR_kernel(const float* __restrict__ A, float* __restrict__ R, float invm) {
    int idx = blockIdx.x * 256 + threadIdx.x;
    int rem = idx & 65535;
    R[idx]  = A[idx] * invm + (((rem >> 8) == (rem & 255)) ? 1.0f : 0.0f);
}

// ---------------------------------------------------------------------------
// Batched GEMM on V_WMMA_F32_16X16X4_F32:   C = alpha*(A@B) + addI*I
//   block = 256 thr (8 wave32), C tile 128x64; wave wm owns rows wm*16..+15
//   and all 64 cols (4 accumulators).  K stepped by 16 with LDS double
//   buffering: A panel (128x16, pad 20) prefetched by async global->LDS b128;
//   B panel stored transposed (Bst[col][k], pad 18) so the {k,k+1} fragment
//   pair is one aligned ds_load_b64 (no repack moves).
// Fragment layouts (ISA 7.12.2, 32-bit):
//   A 16x4: lane<16 -> {K0,K1}, lane>=16 -> {K2,K3}, row M=lane&15
//   B 4x16: v0/v1 = rows {K0,K1} (lanes<16) / {K2,K3} (lanes>=16), col N=lane&15
//   C 16x16: vgpr r -> M = r + 8*(lane>>4), N = lane&15
// ---------------------------------------------------------------------------
#define AS_LD 20   // 16 + 4 pad (float4-aligned rows, conflict-free b64 reads)
#define BS_LD 18   // 16 + 2 pad (b64-aligned, conflict-spread reads)

__global__ __launch_bounds__(256)
void gemm_wmma_f32(const float* __restrict__ Ag, const float* __restrict__ Bg,
                   float* __restrict__ Cg,
                   int K, int lda, int ldb, int ldc,
                   long strideA, long strideB, long strideC,
                   float alpha, float addI) {
    __shared__ float As[2][128 * AS_LD];
    __shared__ float Bst[2][64 * BS_LD];

    const float* Ab = Ag + (long)blockIdx.z * strideA;
    const float* Bb = Bg + (long)blockIdx.z * strideB;
    float*       Cb = Cg + (long)blockIdx.z * strideC;

    int mBase = blockIdx.y * 128;
    int nBase = blockIdx.x * 64;
    int tid   = threadIdx.x;
    int wave  = tid >> 5, lane = tid & 31;
    int half  = lane >> 4, l15 = lane & 15;

    // staging coords
    int ar0 = tid >> 2,        ac = (tid & 3) << 2;    // A: rows 0..63 (+64 on 2nd)
    int brr = tid >> 4,        bcc = (tid & 15) << 2;  // B: row 0..15, col base

    v8f acc[4];
#pragma unroll
    for (int st = 0; st < 4; ++st) acc[st] = (v8f){};

    // ---- stage K-panel [kk, kk+16) into buffer `buf` ----
    auto stage = [&](int kk, int buf) {
        // A panel: 2 async 16B lanes per thread (512 float4 total)
        ASYNC_G2L_B128(&As[buf][ar0 * AS_LD + ac],
                       &Ab[(long)(mBase + ar0) * lda + kk + ac]);
        ASYNC_G2L_B128(&As[buf][(ar0 + 64) * AS_LD + ac],
                       &Ab[(long)(mBase + ar0 + 64) * lda + kk + ac]);
        // B panel: coalesced float4 read, transposed scatter into LDS
        float4 bv = *(const float4*)&Bb[(long)(kk + brr) * ldb + nBase + bcc];
        Bst[buf][(bcc + 0) * BS_LD + brr] = bv.x;
        Bst[buf][(bcc + 1) * BS_LD + brr] = bv.y;
        Bst[buf][(bcc + 2) * BS_LD + brr] = bv.z;
        Bst[buf][(bcc + 3) * BS_LD + brr] = bv.w;
    };

    stage(0, 0);
    WAIT_ASYNC0();
    __syncthreads();

    for (int kk = 0; kk < K; kk += 16) {
        int cur = (kk >> 4) & 1;
        if (kk + 16 < K) stage(kk + 16, cur ^ 1);   // prefetch next panel

#pragma unroll
        for (int ks = 0; ks < 4; ++ks) {
            int kfrag = ks * 4 + half * 2;          // {k, k+1} pair base
            v2f a = *(const v2f*)&As[cur][(wave * 16 + l15) * AS_LD + kfrag];
#pragma unroll
            for (int st = 0; st < 4; ++st) {
                v2f b = *(const v2f*)&Bst[cur][(st * 16 + l15) * BS_LD + kfrag];
                acc[st] = __builtin_amdgcn_wmma_f32_16x16x4_f32(
                    false, a, false, b, (short)0, acc[st], false, false);
            }
        }
        WAIT_ASYNC0();       // this wave's prefetch landed
        __syncthreads();     // everyone's panel writes visible / reads retired
    }

    // Epilogue: C = alpha*acc + addI*I (fused Horner identity-add)
    int growBase = mBase + wave * 16 + half * 8;
#pragma unroll
    for (int st = 0; st < 4; ++st) {
        int gcol = nBase + st * 16 + l15;
#pragma unroll
        for (int r = 0; r < 8; ++r) {
            int grow = growBase + r;
            Cb[(long)grow * ldc + gcol] =
                alpha * acc[st][r] + ((grow == gcol) ? addI : 0.0f);
        }
    }
}

// ---------------------------------------------------------------------------
extern "C" void kernel_launch(void* const* d_in, const int* in_sizes, int n_in,
                              void* d_out, int out_size, void* d_ws, size_t ws_size,
                              hipStream_t stream) {
    (void)in_sizes; (void)n_in; (void)out_size; (void)ws_size;

    const float* q     = (const float*)d_in[0];
    const float* k     = (const float*)d_in[1];
    /* d_in[2] = scale : accepted but unused by the reference forward */
    const float* gridf = (const float*)d_in[3];
    const float* bw    = (const float*)d_in[4];
    const float* coef  = (const float*)d_in[5];
    const float* sbase = (const float*)d_in[6];
    const float* ssp   = (const float*)d_in[7];
    const float* lw    = (const float*)d_in[8];
    const float* lb    = (const float*)d_in[9];

    float* ws   = (float*)d_ws;
    float* sq   = ws;                        // 49152
    float* sk   = ws + BHP;                  // 49152
    float* rlw  = ws + 2 * BHP;              // 256
    float* skw  = ws + 2 * BHP + PP;         // 49152
    float* Abuf = ws + 3 * BHP + PP;         // 192*65536 (offset 147712, aligned)
    float* Rbuf = Abuf + TOTEL;
    float* Tbuf = Rbuf + TOTEL;

    // (1) feature map + sigmoid row sums for q and k
    sigrow_kernel<<<2 * BHP, 64, 0, stream>>>(q, k, gridf, bw, coef, sbase, ssp, sq, sk);

    // (2) skw (192 GEMVs) + rlw (row sums of lin_w)
    skw_kernel<<<BH + 1, 256, 0, stream>>>(sk, lw, skw, rlw);

    // (3) scaled expm argument:  A = 2^-s * (f_out @ lin_w^T + lin_b)
    const int   S    = 10;               // ||A||_1 ~ 1e2 -> ||A/2^s|| <~ 0.15
    const int   MDEG = 8;                // trunc err ~ 0.15^9/9! << fp32 eps
    buildA_kernel<<<TOTEL / 256, 256, 0, stream>>>(sq, rlw, skw, lb, Abuf, 1.0f / 1024.0f);

    // (4) Horner:  R = I + A/m;  for j=m-1..1: R = I + (A@R)/j
    initR_kernel<<<TOTEL / 256, 256, 0, stream>>>(Abuf, Rbuf, 1.0f / (float)MDEG);

    float* Rp = Rbuf;
    float* Tp = Tbuf;
    dim3 gPP(4, 2, BH);   // 64-col x 128-row tiles over 256x256, 192 matrices
    for (int j = MDEG - 1; j >= 1; --j) {
        gemm_wmma_f32<<<gPP, 256, 0, stream>>>(Abuf, Rp, Tp, PP, PP, PP, PP,
                                               (long)MATEL, (long)MATEL, (long)MATEL,
                                               1.0f / (float)j, 1.0f);
        float* t = Rp; Rp = Tp; Tp = t;
    }

    // (5) s squarings:  R <- R@R
    for (int sIt = 0; sIt < S; ++sIt) {
        gemm_wmma_f32<<<gPP, 256, 0, stream>>>(Rp, Rp, Tp, PP, PP, PP, PP,
                                               (long)MATEL, (long)MATEL, (long)MATEL,
                                               1.0f, 0.0f);
        float* t = Rp; Rp = Tp; Tp = t;
    }

    // (6) outputs: q passthrough + k_new = E @ k
    float* outq = (float*)d_out;
    float* outk = outq + QEL;
    hipMemcpyAsync(outq, q, (size_t)QEL * sizeof(float),
                   hipMemcpyDeviceToDevice, stream);

    dim3 gEK(1, 2, BH);   // C is 256x64 per bh
    gemm_wmma_f32<<<gEK, 256, 0, stream>>>(Rp, k, outk, PP, PP, DD, DD,
                                           (long)MATEL, (long)(PP * DD), (long)(PP * DD),
                                           1.0f, 0.0f);
}